// Layer_C3_7292854468919
// MI455X (gfx1250) — compile-verified
//
#include <hip/hip_runtime.h>

typedef __attribute__((ext_vector_type(2))) float v2f;
typedef __attribute__((ext_vector_type(4))) float v4f;
typedef __attribute__((ext_vector_type(8))) float v8f;
typedef unsigned int v4u __attribute__((ext_vector_type(4)));
typedef int v8i_t __attribute__((ext_vector_type(8)));
typedef int v4i_t __attribute__((ext_vector_type(4)));

// LDS tile layout [ci][row][col] (matches TDM fill order: X=col fastest, then
// Y=row, then Z=ci). Strides in floats:
#define SROW 256          // per input row
#define SCI  2048         // per input channel = 8 rows * 256

__device__ const int d_conn[6][10] = {
    {0, 4, 5, 6, 9, 10, 11, 12, 14, 15},
    {0, 1, 5, 6, 7, 10, 11, 12, 13, 15},
    {0, 1, 2, 6, 7, 8, 11, 13, 14, 15},
    {1, 2, 3, 6, 7, 8, 9, 12, 14, 15},
    {2, 3, 4, 7, 8, 9, 10, 12, 13, 15},
    {3, 4, 5, 8, 9, 10, 11, 13, 14, 15}};

// ---------------------------------------------------------------------------
// Prep: scatter weight[10,6,5,5] through the C3 table into per-lane WMMA B
// fragments (38 chunks x 32 lanes x 2 floats) + effective bias (bias*count).
// Chunk -> tap map (must match A addressing in main kernel):
//   c<36: ci=c/6, s=c%6; s<5 : K -> (ci, kr=s, kc=K)
//                         s==5: K -> (ci, kr=K, kc=4)
//   c==36:                K -> (ci=K, kr=4, kc=4)
//   c==37:                K=0 ->(4,4,4), K=1 ->(5,4,4), K>=2 -> zero pad
// B lane layout (mirrors f32 16x16x4 A layout): lane l holds N = l&15,
// VGPR0 = B[K=2*(l>=16)][N], VGPR1 = B[K+1][N].
// ---------------------------------------------------------------------------
__global__ void c3_prep_kernel(const float* __restrict__ weight,
                               const float* __restrict__ bias,
                               float* __restrict__ wsB,      // 38*32*2 floats
                               float* __restrict__ biasEff)  // 16 floats
{
    int tid = blockIdx.x * blockDim.x + threadIdx.x;
    if (tid < 38 * 32) {
        int c  = tid >> 5;
        int l  = tid & 31;
        int n  = l & 15;
        int kl = (l >> 4) * 2;
        float vals[2];
        for (int q = 0; q < 2; ++q) {
            int K = kl + q;
            int ci = 0, kr = 0, kc = 0;
            bool valid = true;
            if (c < 36) {
                int cci = c / 6, s = c % 6;
                if (s < 5) { ci = cci; kr = s; kc = K; }
                else       { ci = cci; kr = K; kc = 4; }
            } else if (c == 36) {
                ci = K; kr = 4; kc = 4;
            } else {
                if (K == 0)      { ci = 4; kr = 4; kc = 4; }
                else if (K == 1) { ci = 5; kr = 4; kc = 4; }
                else             valid = false;
            }
            float w = 0.0f;
            if (valid) {
                int j = -1;
                for (int jj = 0; jj < 10; ++jj)
                    if (d_conn[ci][jj] == n) j = jj;
                if (j >= 0) w = weight[((j * 6 + ci) * 5 + kr) * 5 + kc];
            }
            vals[q] = w;
        }
        wsB[(c * 32 + l) * 2 + 0] = vals[0];
        wsB[(c * 32 + l) * 2 + 1] = vals[1];
    }
    if (tid >= 38 * 32 && tid < 38 * 32 + 16) {
        int oc = tid - 38 * 32;
        int cnt = 0;
        for (int ci = 0; ci < 6; ++ci)
            for (int j = 0; j < 10; ++j)
                if (d_conn[ci][j] == oc) ++cnt;
        biasEff[oc] = bias[oc] * (float)cnt;
    }
}

// ---------------------------------------------------------------------------
// Main: implicit-GEMM conv via V_WMMA_F32_16X16X4_F32.
// Block = 256 threads (8 wave32). Block tile: batch n, 4 output rows, all
// 252 cols. The 49KB input tile (6ch x 8 rows x 256 cols) is DMA'd into LDS
// by ONE Tensor Data Mover descriptor issued from wave 0 (tracked by
// TENSORcnt), then 8 waves run 38 WMMAs per 16-pixel M-tile over K=150 with
// B fragments held in registers. Output stored non-temporally (write-only
// stream; keep the input resident in the 192MB L2 instead).
// ---------------------------------------------------------------------------
__launch_bounds__(256)
__global__ void c3_wmma_kernel(const float* __restrict__ x,
                               const float* __restrict__ wsB,
                               const float* __restrict__ biasEff,
                               float* __restrict__ out)
{
    __shared__ float sm[6 * SCI];  // 49152 bytes, [ci][row][col]

    const int n  = blockIdx.y;
    const int r0 = blockIdx.x * 4;          // first output row of this block
    const int tid = threadIdx.x;
    const int wave = tid >> 5;

    // ---- TDM: DMA input tile x[n, 0:6, r0:r0+8, 0:256] -> LDS ----
    if (wave == 0) {
        const float* xt = x + ((size_t)n * 6 * 256 + (size_t)r0) * 256;
        unsigned long long ga = (unsigned long long)(uintptr_t)xt;
        unsigned int lds_base = (unsigned int)(uintptr_t)(&sm[0]);

        v4u g0;
        g0[0] = 1u;                               // count=1, user descriptor
        g0[1] = lds_base;                         // lds_addr (bytes)
        g0[2] = (unsigned int)ga;                 // global_addr[31:0]
        g0[3] = (unsigned int)((ga >> 32) & 0x01FFFFFFu) | (2u << 30);  // [56:32] | type=2

        v8i_t g1;
        g1[0] = (int)(2u << 16);                  // data_size=2 (4 bytes)
        g1[1] = (int)(256u << 16);                // tensor_dim0[15:0]=256
        g1[2] = (int)(256u << 16);                // dim0 hi=0 | tensor_dim1[15:0]=256
        g1[3] = (int)(256u << 16);                // dim1 hi=0 | tile_dim0=256
        g1[4] = (int)(8u | (6u << 16));           // tile_dim1=8 | tile_dim2=6
        g1[5] = (int)256u;                        // tensor_dim0_stride=256 (row pitch)
        g1[6] = 0;                                // stride0 hi | tensor_dim1_stride lo16=0
        g1[7] = 1;                                // tensor_dim1_stride[47:16]: 65536>>16
        v4i_t g2;
        g2[0] = 6;                                // tensor_dim2 = 6 channels
        g2[1] = 0; g2[2] = 0; g2[3] = 0;
        v4i_t g3;
        g3[0] = 0; g3[1] = 0; g3[2] = 0; g3[3] = 0;
        v8i_t g4;                                 // extra operand on clang-23
        g4[0] = 0; g4[1] = 0; g4[2] = 0; g4[3] = 0;
        g4[4] = 0; g4[5] = 0; g4[6] = 0; g4[7] = 0;

        __builtin_amdgcn_tensor_load_to_lds(g0, g1, g2, g3, g4, 0);
        __builtin_amdgcn_s_wait_tensorcnt((short)0);
    }
    __syncthreads();

    const int lane = tid & 31;
    const int half = lane >> 4;   // which 16-lane half (selects K pair / M high)
    const int m    = lane & 15;   // N index (output channel) and A pixel lane
    const int rw   = wave >> 1;   // output row within block tile (0..3)
    const int hw   = wave & 1;    // column half (0: tiles 0-7, 1: tiles 8-15)
    const int y    = r0 + rw;     // global output row (< 252 always)

    // ---- B fragments into registers (reused for all 8 M-tiles) ----
    v2f bfrag[38];
    const v2f* Bp = (const v2f*)wsB;
#pragma unroll
    for (int c = 0; c < 38; ++c) bfrag[c] = Bp[c * 32 + lane];
    const float bv = biasEff[m];

    const int h2 = half * 2;      // K offset for this lane half

#pragma unroll 1
    for (int t = 0; t < 8; ++t) {
        int c0 = (hw * 8 + t) * 16;
        if (c0 > 236) c0 = 236;   // last tile overlaps; writes identical data

        const float* bp  = sm + rw * SROW + c0 + m;  // pixel-lane base
        const float* bpA = bp + h2;                  // + kl in kc
        const float* bpB = bp + h2 * SROW;           // + kl in kr
        const float* bpC = bp + h2 * SCI;            // + kl in ci

        v8f acc;
#pragma unroll
        for (int v = 0; v < 8; ++v) acc[v] = bv;

#pragma unroll
        for (int c = 0; c < 38; ++c) {
            float a0, a1;
            if (c < 36) {
                int ci = c / 6, s = c % 6;
                if (s < 5) {  // taps (ci, kr=s, kc=0..3): contiguous cols
                    const float* p = bpA + ci * SCI + s * SROW;
                    a0 = p[0]; a1 = p[1];
                } else {      // taps (ci, kr=0..3, kc=4): stride = row
                    const float* p = bpB + ci * SCI + 4;
                    a0 = p[0]; a1 = p[SROW];
                }
            } else if (c == 36) {  // taps (ci=0..3, kr=4, kc=4): stride = ch
                const float* p = bpC + 4 * SROW + 4;
                a0 = p[0]; a1 = p[SCI];
            } else {               // taps (ci=4..5, kr=4, kc=4); K>=2 B-zeroed
                const float* p = bp + 4 * SCI + 4 * SROW + 4;
                a0 = p[0]; a1 = p[SCI];
            }
            v2f a; a[0] = a0; a[1] = a1;
            acc = __builtin_amdgcn_wmma_f32_16x16x4_f32(
                false, a, false, bfrag[c], (short)0, acc, false, false);
        }

        // D layout: lane l, VGPR v -> M = v + 8*half, N = l&15
        float* op = out + (((size_t)(n * 16 + m) * 252 + y) * 252) + c0 + half * 8;
        v4f lo, hi;
        lo[0] = acc[0]; lo[1] = acc[1]; lo[2] = acc[2]; lo[3] = acc[3];
        hi[0] = acc[4]; hi[1] = acc[5]; hi[2] = acc[6]; hi[3] = acc[7];
        __builtin_nontemporal_store(lo, (v4f*)op);
        __builtin_nontemporal_store(hi, (v4f*)(op + 4));
    }
}

extern "C" void kernel_launch(void* const* d_in, const int* in_sizes, int n_in,
                              void* d_out, int out_size, void* d_ws, size_t ws_size,
                              hipStream_t stream) {
    const float* x      = (const float*)d_in[0];  // [64,6,256,256]
    const float* weight = (const float*)d_in[1];  // [10,6,5,5]
    const float* bias   = (const float*)d_in[2];  // [1,16,1,1]
    float* out = (float*)d_out;                   // [64,16,252,252]

    float* wsB     = (float*)d_ws;                // 38*32*2 floats
    float* biasEff = wsB + 38 * 32 * 2;           // 16 floats

    c3_prep_kernel<<<dim3(5), dim3(256), 0, stream>>>(weight, bias, wsB, biasEff);

    dim3 grid(63, 64);  // 63 row-blocks (252/4) x 64 batch
    c3_wmma_kernel<<<grid, dim3(256), 0, stream>>>(x, wsB, biasEff, out);
}